// DivEncLayer_61486751809803
// MI455X (gfx1250) — compile-verified
//
#include <hip/hip_runtime.h>

typedef float v2f __attribute__((ext_vector_type(2)));
typedef float v8f __attribute__((ext_vector_type(8)));

constexpr int Q = 128;   // slices
constexpr int S = 8;     // slice length (K of GEMM 1)
constexpr int H = 32;    // hidden units (N of GEMM 1)
constexpr int BT = 16;   // batch rows per block (M of WMMA)
constexpr float BN_EPS = 1e-3f;

// One block: 8 waves, 16 batch rows, all 128 slices (16 q per wave).
// Per q: D(16x32) = A(16x8) @ B(8x32) + b1 via 4x v_wmma_f32_16x16x4_f32,
// with the Dense bias carried in the WMMA C operand (free on the matrix pipe).
// Epilogue folds BN + W2 into one coefficient pair per h:
//   out[M] = sum_h elu[M][h]*c1_h + sum_h c0_h + b2[q]
//   c1_h = gamma*rsqrt(var+eps)*w2 ; c0_h = (beta - mu*inv)*w2
// so each element costs: ELU + 2 FMA.
__global__ __launch_bounds__(256) void divenc_wmma_f32(
    const float* __restrict__ x,     // (B, Q*S)
    const float* __restrict__ W1,    // (Q, S, H)
    const float* __restrict__ b1g,   // (Q, H)
    const float* __restrict__ gam,   // (Q, H)
    const float* __restrict__ bet,   // (Q, H)
    const float* __restrict__ mmean, // (Q, H)
    const float* __restrict__ mvar,  // (Q, H)
    const float* __restrict__ W2,    // (Q, H)
    const float* __restrict__ b2g,   // (Q,)
    float* __restrict__ out)         // (B, Q)
{
    const int lane  = threadIdx.x & 31;
    const int wave  = threadIdx.x >> 5;          // 0..7 -> q block
    const int bbase = blockIdx.x * BT;

    const int mrow  = lane & 15;                 // A: M row (batch)
    const int khalf = lane >> 4;                 // A/B: K half select (0 or 1)
    const int ncol  = lane & 15;                 // B/C/D: N column (h index)

    const float* xrow = x + (size_t)(bbase + mrow) * (Q * S);

    for (int qi = 0; qi < 16; ++qi) {
        const int q = wave * 16 + qi;
        const float* xq = xrow + q * S;

        // A fragments: lane holds row mrow, K = {2*khalf, 2*khalf+1} (+4 for hi)
        v2f a_lo = { xq[2 * khalf + 0], xq[2 * khalf + 1] };   // K = 0..3
        v2f a_hi = { xq[2 * khalf + 4], xq[2 * khalf + 5] };   // K = 4..7

        // B fragments: lane holds col h, K = {2*khalf, 2*khalf+1} (+4 for hi)
        const float* w1q = W1 + q * (S * H);
        const int h0 = ncol, h1 = ncol + 16;
        const int kb = 2 * khalf;
        v2f bt0_lo = { w1q[(kb + 0) * H + h0], w1q[(kb + 1) * H + h0] };
        v2f bt0_hi = { w1q[(kb + 4) * H + h0], w1q[(kb + 5) * H + h0] };
        v2f bt1_lo = { w1q[(kb + 0) * H + h1], w1q[(kb + 1) * H + h1] };
        v2f bt1_hi = { w1q[(kb + 4) * H + h1], w1q[(kb + 5) * H + h1] };

        // Dense bias rides in the WMMA C operand: C/D element (M,N) has
        // N = h fixed per lane, so a per-lane broadcast is the correct bias.
        const int i0 = q * H + h0, i1 = q * H + h1;
        const float bias0 = b1g[i0], bias1 = b1g[i1];
        v8f acc0 = { bias0, bias0, bias0, bias0, bias0, bias0, bias0, bias0 };
        v8f acc1 = { bias1, bias1, bias1, bias1, bias1, bias1, bias1, bias1 };

        // D = A x B + C, chained over K=8 in two x4 steps
        acc0 = __builtin_amdgcn_wmma_f32_16x16x4_f32(false, a_lo, false, bt0_lo,
                                                     (short)0, acc0, false, false);
        acc0 = __builtin_amdgcn_wmma_f32_16x16x4_f32(false, a_hi, false, bt0_hi,
                                                     (short)0, acc0, false, false);
        acc1 = __builtin_amdgcn_wmma_f32_16x16x4_f32(false, a_lo, false, bt1_lo,
                                                     (short)0, acc1, false, false);
        acc1 = __builtin_amdgcn_wmma_f32_16x16x4_f32(false, a_hi, false, bt1_hi,
                                                     (short)0, acc1, false, false);

        // Per-lane scalar coefficients (h fixed per lane per tile), amortized 8x.
        // Raw v_rsq_f32: mov_var + eps is ~[1.0, 2.1], no denorm guard needed.
        const float inv0 = gam[i0] * __builtin_amdgcn_rsqf(mvar[i0] + BN_EPS);
        const float inv1 = gam[i1] * __builtin_amdgcn_rsqf(mvar[i1] + BN_EPS);
        const float w2_0 = W2[i0], w2_1 = W2[i1];
        const float c1_0 = inv0 * w2_0;
        const float c1_1 = inv1 * w2_1;
        const float c0s  = fmaf(-mmean[i0] * inv0 + bet[i0], w2_0,
                                (-mmean[i1] * inv1 + bet[i1]) * w2_1);

        float part[8];
        #pragma unroll
        for (int v = 0; v < 8; ++v) {
            float t0 = acc0[v];
            t0 = t0 > 0.0f ? t0 : (__expf(t0) - 1.0f);      // ELU
            float t1 = acc1[v];
            t1 = t1 > 0.0f ? t1 : (__expf(t1) - 1.0f);
            // BN + Dense(1) folded: 2 FMAs; constant term rides in c0s,
            // summed exactly once per output by the 16-lane reduction.
            part[v] = fmaf(t0, c1_0, fmaf(t1, c1_1, c0s));
        }

        // Reduce over the 16 lanes of each half-wave (sum over h).
        #pragma unroll
        for (int off = 8; off >= 1; off >>= 1) {
            #pragma unroll
            for (int v = 0; v < 8; ++v)
                part[v] += __shfl_xor(part[v], off, 32);
        }

        // Lane 0 holds out rows M=0..7, lane 16 holds M=8..15.
        if (ncol == 0) {
            const float bq = b2g[q];
            float* op = out + (size_t)(bbase + khalf * 8) * Q + q;
            #pragma unroll
            for (int v = 0; v < 8; ++v)
                op[(size_t)v * Q] = part[v] + bq;
        }
    }
}

extern "C" void kernel_launch(void* const* d_in, const int* in_sizes, int n_in,
                              void* d_out, int out_size, void* d_ws, size_t ws_size,
                              hipStream_t stream) {
    const float* x     = (const float*)d_in[0];
    const float* W1    = (const float*)d_in[1];
    const float* b1    = (const float*)d_in[2];
    const float* gamma = (const float*)d_in[3];
    const float* beta  = (const float*)d_in[4];
    const float* mmean = (const float*)d_in[5];
    const float* mvar  = (const float*)d_in[6];
    const float* W2    = (const float*)d_in[7];
    const float* b2    = (const float*)d_in[8];
    float* out = (float*)d_out;

    const int B = in_sizes[0] / (Q * S);   // 32768
    dim3 grid(B / BT);                     // 2048 blocks
    dim3 block(256);                       // 8 waves (wave32)
    divenc_wmma_f32<<<grid, block, 0, stream>>>(x, W1, b1, gamma, beta,
                                                mmean, mvar, W2, b2, out);
}